// SOM_23115513987478
// MI455X (gfx1250) — compile-verified
//
#include <hip/hip_runtime.h>
#include <hip/hip_bf16.h>
#include <stdint.h>

typedef __attribute__((ext_vector_type(2))) float v2f;
typedef __attribute__((ext_vector_type(8))) float v8f;

#define MAP_ROWS 16384   // somap rows (xs*ys)
#define NPTS     8192    // x rows
#define DIM      64
#define XS       128

#define WM 2             // m-tiles per wave
#define WN 4             // n-tiles per wave
// Block: 8 waves = 4 (M) x 2 (N) -> covers 128 rows x 128 cols of dists.

// ---------------------------------------------------------------------------
// Kernel 1: init argmin keys to +inf
// ---------------------------------------------------------------------------
__global__ void som_init_keys(unsigned long long* __restrict__ keys) {
    int n = blockIdx.x * blockDim.x + threadIdx.x;
    if (n < NPTS) keys[n] = 0xFFFFFFFFFFFFFFFFull;
}

// ---------------------------------------------------------------------------
// Kernel 2: row squared norms for somap and x
// ---------------------------------------------------------------------------
__global__ void som_norms(const float* __restrict__ somap,
                          const float* __restrict__ xm,
                          float* __restrict__ nmap,
                          float* __restrict__ nx) {
    int tid = blockIdx.x * blockDim.x + threadIdx.x;
    const float* row;
    float* dst;
    int idx;
    if (tid < MAP_ROWS) {
        idx = tid; row = somap + (size_t)idx * DIM; dst = nmap;
    } else if (tid < MAP_ROWS + NPTS) {
        idx = tid - MAP_ROWS; row = xm + (size_t)idx * DIM; dst = nx;
    } else {
        return;
    }
    float s = 0.0f;
    #pragma unroll
    for (int i = 0; i < DIM / 4; ++i) {
        float4 v = ((const float4*)row)[i];
        s += v.x * v.x + v.y * v.y + v.z * v.z + v.w * v.w;
    }
    dst[idx] = s;
}

// ---------------------------------------------------------------------------
// Kernel 3: register-blocked WMMA f32 kernel.
//   Each wave32 computes WM x WN = 2x4 tiles (32 rows x 64 cols) with
//   8 f32 accumulators, 128 V_WMMA_F32_16X16X4_F32 per wave.
//   A fragments reused WN times, B fragments WM times from registers
//   => L2 read traffic drops ~2.7x vs one-tile-per-wave.
//   Fused per-column argmin via packed u64 atomicMin; NT streaming stores.
// ---------------------------------------------------------------------------
__global__ __launch_bounds__(256) void som_dist_kernel(
        const float* __restrict__ somap,
        const float* __restrict__ xm,
        const float* __restrict__ nmap,
        const float* __restrict__ nx,
        float* __restrict__ dists,
        unsigned long long* __restrict__ keys) {
    const int lane = threadIdx.x & 31;
    const int wave = threadIdx.x >> 5;
    const int r    = lane & 15;     // M for A-lanes, N for B/C/D-lanes
    const int half = lane >> 4;     // A: K-pair select; C/D: M = j + 8*half

    const int waveM = wave >> 1;    // 0..3
    const int waveN = wave & 1;     // 0..1

    // Tile bases
    int m0[WM], n0[WN];
    #pragma unroll
    for (int mi = 0; mi < WM; ++mi)
        m0[mi] = (blockIdx.x * (4 * WM) + waveM * WM + mi) * 16;
    #pragma unroll
    for (int ni = 0; ni < WN; ++ni)
        n0[ni] = (blockIdx.y * (2 * WN) + waveN * WN + ni) * 16;

    // Per-lane source rows (A layout 16x4 f32 / B layout 4x16 f32, wave32)
    const float* arow[WM];
    const float* brow[WN];
    #pragma unroll
    for (int mi = 0; mi < WM; ++mi)
        arow[mi] = somap + (size_t)(m0[mi] + r) * DIM + 2 * half;
    #pragma unroll
    for (int ni = 0; ni < WN; ++ni)
        brow[ni] = xm + (size_t)(n0[ni] + r) * DIM + 2 * half;

    v8f acc[WM][WN];
    #pragma unroll
    for (int mi = 0; mi < WM; ++mi)
        #pragma unroll
        for (int ni = 0; ni < WN; ++ni)
            acc[mi][ni] = (v8f){};

    #pragma unroll
    for (int kk = 0; kk < DIM / 4; ++kk) {
        v2f a[WM], b[WN];
        #pragma unroll
        for (int mi = 0; mi < WM; ++mi)
            a[mi] = *(const v2f*)(arow[mi] + kk * 4);
        #pragma unroll
        for (int ni = 0; ni < WN; ++ni)
            b[ni] = *(const v2f*)(brow[ni] + kk * 4);
        #pragma unroll
        for (int mi = 0; mi < WM; ++mi)
            #pragma unroll
            for (int ni = 0; ni < WN; ++ni)
                acc[mi][ni] = __builtin_amdgcn_wmma_f32_16x16x4_f32(
                        false, a[mi], false, b[ni],
                        (short)0, acc[mi][ni], false, false);
    }

    // Hoist map-row norms: depend only on (mi, half, j), reused for all ni.
    float a2v[WM][8];
    #pragma unroll
    for (int mi = 0; mi < WM; ++mi)
        #pragma unroll
        for (int j = 0; j < 8; ++j)
            a2v[mi][j] = nmap[m0[mi] + 8 * half + j];

    #pragma unroll
    for (int ni = 0; ni < WN; ++ni) {
        const int n  = n0[ni] + r;
        const float b2 = nx[n];
        float bestD = 3.402823466e38f;
        int   bestM = 0;
        #pragma unroll
        for (int mi = 0; mi < WM; ++mi) {
            #pragma unroll
            for (int j = 0; j < 8; ++j) {
                const int m = m0[mi] + 8 * half + j;
                float d2 = a2v[mi][j] + b2 - 2.0f * acc[mi][ni][j];
                d2 = fmaxf(d2, 0.0f);
                // streaming store: 536 MB >> 192 MB L2, keep inputs resident
                __builtin_nontemporal_store(sqrtf(d2),
                                            dists + (size_t)m * NPTS + n);
                // m ascending + strict '<' => ties keep the lowest index
                if (d2 < bestD) { bestD = d2; bestM = m; }
            }
        }

        // Pack (dist_bits, index): non-negative f32 bits order-preserve as u32;
        // u64 min breaks distance ties toward the lowest map index (jnp.argmin).
        unsigned long long key =
            ((unsigned long long)__float_as_uint(bestD) << 32) | (unsigned)bestM;

        // Combine the two column halves (lanes r and r+16) on wave32.
        unsigned lo  = (unsigned)key;
        unsigned hi  = (unsigned)(key >> 32);
        unsigned olo = (unsigned)__shfl_xor((int)lo, 16, 32);
        unsigned ohi = (unsigned)__shfl_xor((int)hi, 16, 32);
        unsigned long long other = ((unsigned long long)ohi << 32) | olo;
        if (other < key) key = other;

        if (half == 0) {
            atomicMin(&keys[n], key);   // global_atomic_min_u64
        }
    }
}

// ---------------------------------------------------------------------------
// Kernel 4: keys -> (bmu_x, bmu_y) coords
// ---------------------------------------------------------------------------
__global__ void som_finalize(const unsigned long long* __restrict__ keys,
                             float* __restrict__ out) {
    int n = blockIdx.x * blockDim.x + threadIdx.x;
    if (n >= NPTS) return;
    unsigned m = (unsigned)(keys[n] & 0xFFFFFFFFu);
    out[2 * n + 0] = (float)(m / XS);
    out[2 * n + 1] = (float)(m % XS);
}

// ---------------------------------------------------------------------------
// Launch
// ---------------------------------------------------------------------------
extern "C" void kernel_launch(void* const* d_in, const int* in_sizes, int n_in,
                              void* d_out, int out_size, void* d_ws, size_t ws_size,
                              hipStream_t stream) {
    const float* somap = (const float*)d_in[0];   // (16384, 64)
    const float* xm    = (const float*)d_in[1];   // (8192, 64)

    float* out   = (float*)d_out;
    float* dists = out + 2 * NPTS;                // dists after (N,2) coords

    // Workspace layout
    unsigned long long* keys = (unsigned long long*)d_ws;                    // 64 KB
    float* nmap = (float*)((char*)d_ws + NPTS * sizeof(unsigned long long)); // 64 KB
    float* nx   = nmap + MAP_ROWS;                                           // 32 KB

    som_init_keys<<<(NPTS + 255) / 256, 256, 0, stream>>>(keys);

    som_norms<<<(MAP_ROWS + NPTS + 255) / 256, 256, 0, stream>>>(somap, xm, nmap, nx);

    // Block covers 128x128 of dists; grid = (16384/128, 8192/128)
    dim3 grid(MAP_ROWS / 128, NPTS / 128);
    som_dist_kernel<<<grid, 256, 0, stream>>>(somap, xm, nmap, nx, dists, keys);

    som_finalize<<<(NPTS + 255) / 256, 256, 0, stream>>>(keys, out);
}